// SupervisedMaskLoss_15762529976902
// MI455X (gfx1250) — compile-verified
//
#include <hip/hip_runtime.h>
#include <math.h>

// Problem constants (match reference)
#define B_  4
#define N_  100000
#define K_  16
#define EPS_ 1e-6f
#define W_CE_ 2.0f
#define W_DICE_ 0.1f

#define BLOCKS_PER_B 128       // blocks per batch in pass1 (256 thr = 8 waves each)
#define CHUNKS (N_ / 4)        // N divisible by 4

// ln2 split for accurate log2 -> ln conversion (2 VALU ops after v_log_f32)
#define LN2_HI 0x1.62e42ep-1f
#define LN2_LO 0x1.efa39ep-25f

typedef __attribute__((ext_vector_type(2))) float v2f;
typedef __attribute__((ext_vector_type(8))) float v8f;

__device__ __forceinline__ v8f wmma_f32_16x16x4(v2f a, v2f b, v8f c) {
    // 8 args: (neg_a, A, neg_b, B, c_mod, C, reuse_a, reuse_b)
    return __builtin_amdgcn_wmma_f32_16x16x4_f32(false, a, false, b, (short)0, c,
                                                 false, false);
}

// -ln(x) via raw v_log_f32 (log2) + hi/lo ln2 multiply. ~1-2 ulp, plenty for a
// loss reduced over 1.6M terms; avoids the ~20-op accurate ocml expansion.
__device__ __forceinline__ float neg_ln(float x) {
    const float l2 = __builtin_amdgcn_logf(x);          // v_log_f32: log2(x)
    return -fmaf(l2, LN2_HI, l2 * LN2_LO);
}

// ---------------------------------------------------------------------------
// Pass 1: accumulate per-batch 16x16 matrices
//   POS  = sum_n v * (-log m)[n,i]     * gt[n,j]
//   NEGG = sum_n v * (-log(1-m))[n,i]  * gt[n,j]
//   INTER= sum_n v * m[n,i]            * gt[n,j]
// plus sum_n v per batch. acc layout: [b][3][16][16] floats, then vsum[b].
// ---------------------------------------------------------------------------
__global__ __launch_bounds__(256) void
mask_loss_pass1(const float* __restrict__ mask, const float* __restrict__ gt,
                const float* __restrict__ valid, float* __restrict__ acc) {
    __shared__ float sred[3 * 256];
    __shared__ float svsum;

    const int t = threadIdx.x;
    for (int i = t; i < 3 * 256; i += 256) sred[i] = 0.f;
    if (t == 0) svsum = 0.f;
    __syncthreads();

    const int lane = t & 31;
    const int wave = t >> 5;
    const int hi   = lane >> 4;   // 0: lanes 0-15, 1: lanes 16-31
    const int l16  = lane & 15;   // column index
    const int b    = blockIdx.y;
    const size_t base = (size_t)b * N_;

    v8f cpos = {0.f, 0.f, 0.f, 0.f, 0.f, 0.f, 0.f, 0.f};
    v8f cneg = cpos;
    v8f cint = cpos;
    float vsum = 0.f;

    const int wgid   = blockIdx.x * 8 + wave;
    const int stride = gridDim.x * 8;

#pragma unroll 2
    for (int c = wgid; c < CHUNKS; c += stride) {
        // This lane covers rows rA, rA+1 in column l16 for both A and B operands.
        const int rA = c * 4 + 2 * hi;
        const size_t off = (base + rA) * (size_t)K_ + l16;

        // Prefetch the next chunk this wave will touch (strided, cold lines).
        {
            const int cn = (c + stride < CHUNKS) ? (c + stride) : c;
            const size_t offn = (base + cn * 4 + 2 * hi) * (size_t)K_ + l16;
            __builtin_prefetch(&mask[offn], 0, 0);   // global_prefetch_b8
            __builtin_prefetch(&gt[offn], 0, 0);
        }

        const float m0 = mask[off];
        const float m1 = mask[off + K_];
        const float g0 = gt[off];
        const float g1 = gt[off + K_];
        const float v0 = valid[base + rA];
        const float v1 = valid[base + rA + 1];

        const float mc0 = fminf(fmaxf(m0, EPS_), 1.f - EPS_);
        const float mc1 = fminf(fmaxf(m1, EPS_), 1.f - EPS_);
        const float nlm0  = neg_ln(mc0);
        const float nlm1  = neg_ln(mc1);
        const float nl1m0 = neg_ln(1.f - mc0);
        const float nl1m1 = neg_ln(1.f - mc1);

        v2f bg   = {g0, g1};
        v2f apos = {v0 * nlm0,  v1 * nlm1};
        v2f aneg = {v0 * nl1m0, v1 * nl1m1};
        v2f aint = {v0 * m0,    v1 * m1};

        cpos = wmma_f32_16x16x4(apos, bg, cpos);
        cneg = wmma_f32_16x16x4(aneg, bg, cneg);
        cint = wmma_f32_16x16x4(aint, bg, cint);

        vsum += (l16 == 0) ? (v0 + v1) : 0.f;   // branchless, keep EXEC full
    }

    // C layout: reg r holds element (i = r + 8*hi, j = l16)
    for (int r = 0; r < 8; ++r) {
        const int ij = (r + 8 * hi) * 16 + l16;
        atomicAdd(&sred[0 * 256 + ij], cpos[r]);   // ds_add_f32
        atomicAdd(&sred[1 * 256 + ij], cneg[r]);
        atomicAdd(&sred[2 * 256 + ij], cint[r]);
    }
    if (l16 == 0) atomicAdd(&svsum, vsum);
    __syncthreads();

    float* accb = acc + (size_t)b * (3 * 256);
    for (int i = t; i < 3 * 256; i += 256) unsafeAtomicAdd(&accb[i], sred[i]);
    if (t == 0) unsafeAtomicAdd(&acc[B_ * 3 * 256 + b], svsum);
}

// ---------------------------------------------------------------------------
// Hungarian (Jonker-Volgenant, e-maxx variant) on a 16x16 cost, minimize.
// colOut[i] = assigned column for row i (matches scipy linear_sum_assignment).
// ---------------------------------------------------------------------------
__device__ void hungarian16(const float* a, int* colOut) {
    const int n = 16;
    float u[17], v[17], minv[17];
    int p[17], way[17];
    bool used[17];
    for (int i = 0; i <= n; ++i) { u[i] = 0.f; v[i] = 0.f; p[i] = 0; way[i] = 0; }
    for (int i = 1; i <= n; ++i) {
        p[0] = i;
        int j0 = 0;
        for (int j = 0; j <= n; ++j) { minv[j] = 3.4e38f; used[j] = false; }
        do {
            used[j0] = true;
            const int i0 = p[j0];
            int j1 = 0;
            float delta = 3.4e38f;
            for (int j = 1; j <= n; ++j) {
                if (!used[j]) {
                    const float cur = a[(i0 - 1) * 16 + (j - 1)] - u[i0] - v[j];
                    if (cur < minv[j]) { minv[j] = cur; way[j] = j0; }
                    if (minv[j] < delta) { delta = minv[j]; j1 = j; }
                }
            }
            for (int j = 0; j <= n; ++j) {
                if (used[j]) { u[p[j]] += delta; v[j] -= delta; }
                else         { minv[j] -= delta; }
            }
            j0 = j1;
        } while (p[j0] != 0);
        do { const int j1 = way[j0]; p[j0] = p[j1]; j0 = j1; } while (j0);
    }
    for (int j = 1; j <= n; ++j)
        if (p[j] > 0) colOut[p[j] - 1] = j - 1;
}

// ---------------------------------------------------------------------------
// Pass 2 (single block): build costs, match, assemble scalar loss.
// ---------------------------------------------------------------------------
__global__ __launch_bounds__(256) void
mask_loss_pass2(const float* __restrict__ acc, float* __restrict__ out) {
    __shared__ float sPOS[B_][256], sNEG[B_][256], sINT[B_][256];
    __shared__ float sCost[B_][256];
    __shared__ float sSM[B_][16], sSG[B_][16], sS1M[B_][16];
    __shared__ float sSumV[B_], sTot[B_];
    __shared__ int   sCol[B_][16];
    __shared__ float sCE[64], sDI[64];

    const int t = threadIdx.x;

    for (int idx = t; idx < B_ * 256; idx += 256) {
        const int b = idx >> 8, e = idx & 255;
        sPOS[b][e] = acc[b * 768 + 0 * 256 + e];
        sNEG[b][e] = acc[b * 768 + 1 * 256 + e];
        sINT[b][e] = acc[b * 768 + 2 * 256 + e];
    }
    __syncthreads();

    if (t < B_ * 16) {
        const int b = t >> 4, x = t & 15;
        float s1 = 0.f, sm = 0.f, sg = 0.f;
        for (int k = 0; k < 16; ++k) {
            s1 += sNEG[b][x * 16 + k];
            sm += sINT[b][x * 16 + k];
            sg += sINT[b][k * 16 + x];
        }
        sS1M[b][x] = s1; sSM[b][x] = sm; sSG[b][x] = sg;
    }
    __syncthreads();

    if (t < B_) {
        sSumV[t] = acc[B_ * 768 + t];
        float tn = 0.f;
        for (int i = 0; i < 16; ++i) tn += sS1M[t][i];
        sTot[t] = tn;
    }
    __syncthreads();

    for (int idx = t; idx < B_ * 256; idx += 256) {
        const int b = idx >> 8, i = (idx >> 4) & 15, j = idx & 15;
        const float denom = fmaxf(sSumV[b], 1.f);
        const float ce = (sPOS[b][i * 16 + j] + sS1M[b][i] - sNEG[b][i * 16 + j]) / denom;
        const float dice = 1.f - 2.f * sINT[b][i * 16 + j] /
                                     (sSM[b][i] + sSG[b][j] + EPS_);
        sCost[b][i * 16 + j] = W_CE_ * ce + W_DICE_ * dice;
    }
    __syncthreads();

    if (t < B_) hungarian16(&sCost[t][0], &sCol[t][0]);
    __syncthreads();

    if (t < B_ * 16) {
        const int b = t >> 4, i = t & 15;
        const int j = sCol[b][i];
        sCE[t] = sPOS[b][i * 16 + j] - sNEG[b][i * 16 + j];
        sDI[t] = 1.f - 2.f * sINT[b][i * 16 + j] /
                           (sSM[b][i] + sSG[b][j] + EPS_);
    }
    __syncthreads();

    if (t == 0) {
        float ceSum = 0.f, diSum = 0.f, totN = 0.f, sumV = 0.f;
        for (int k = 0; k < B_ * 16; ++k) { ceSum += sCE[k]; diSum += sDI[k]; }
        for (int b = 0; b < B_; ++b) { totN += sTot[b]; sumV += sSumV[b]; }
        const float l_ce   = (ceSum + totN) / (fmaxf(sumV, 1.f) * (float)K_);
        const float l_dice = diSum / (float)(B_ * K_);
        out[0] = W_CE_ * l_ce + W_DICE_ * l_dice;
    }
}

// ---------------------------------------------------------------------------
extern "C" void kernel_launch(void* const* d_in, const int* in_sizes, int n_in,
                              void* d_out, int out_size, void* d_ws, size_t ws_size,
                              hipStream_t stream) {
    const float* mask  = (const float*)d_in[0];   // (B,N,K) f32
    const float* gt    = (const float*)d_in[1];   // (B,N,K) f32 one-hot
    const float* valid = (const float*)d_in[2];   // (B,N)   f32
    float* out = (float*)d_out;
    float* acc = (float*)d_ws;                    // [B][3][256] + vsum[B]

    const size_t acc_bytes = (size_t)(B_ * 3 * 256 + B_) * sizeof(float);
    hipMemsetAsync(d_ws, 0, acc_bytes, stream);   // capture-safe

    dim3 grid(BLOCKS_PER_B, B_);
    mask_loss_pass1<<<grid, 256, 0, stream>>>(mask, gt, valid, acc);
    mask_loss_pass2<<<1, 256, 0, stream>>>(acc, out);
}